// ClothMaskPooling_41695542510189
// MI455X (gfx1250) — compile-verified
//
#include <hip/hip_runtime.h>
#include <hip/hip_bf16.h>
#include <stdint.h>

// ---------------------------------------------------------------------------
// ClothMaskPooling for MI455X (gfx1250)
//   x    : [64, 4096, 24, 12] f32   (B=64, C=4096, HW=288)  ~302 MB -> HBM bound
//   mask : [32, 384, 192]     i32   (tiled x2 along batch)
//   out  : id_feat [64,4096] f32 ++ cloth_feat [64,4096] f32
//
// Roofline: 302 MB / 23.3 TB/s ~= 13.4 us; compute is negligible, so the
// kernel is built around the CDNA5 async data path:
//   * global_load_async_to_lds_b128 (ASYNCcnt) streams x directly into LDS
//     with th:TH_LOAD_NT (read-once stream, don't pollute the 192 MB L2)
//   * each wave stages ONLY its own 4-channel sub-tile and waits on its own
//     ASYNCcnt -> no block-wide barrier at all
//   * wave32 masked max/sum with __shfl_xor butterflies; mask bits are
//     precomputed so bit position == lane id
// ---------------------------------------------------------------------------

#define BATCH 64
#define CHAN  4096
#define HW    288           // 24 * 12
#define MH    384
#define MW    192
#define CH_PER_BLOCK 32     // 8 waves x 4 channels; 36 KB LDS tile
#define NWORDS 9            // 288 mask bits = 9 x u32

// ------------------------------ Phase 1 ------------------------------------
__global__ void cmp_mask_words_kernel(const int* __restrict__ mask,
                                      unsigned* __restrict__ idw,
                                      unsigned* __restrict__ clw,
                                      float* __restrict__ idc,
                                      float* __restrict__ clc) {
    __shared__ int patchCnt[HW];
    __shared__ unsigned fm[NWORDS];
    const int b   = blockIdx.x;
    const int bm  = (b < 32) ? b : b - 32;          // concat([mask,mask],axis=0)
    const int tid = threadIdx.x;
    const int* mbase = mask + (size_t)bm * MH * MW;

    // per-patch count of (m==2 || m==3) over the 16x16 patch
    for (int p = tid; p < HW; p += 256) {
        const int h = p / 12, w = p % 12;
        int cnt = 0;
        for (int ph = 0; ph < 16; ++ph) {
            const int* row = mbase + (h * 16 + ph) * MW + w * 16;
#pragma unroll
            for (int pw = 0; pw < 16; ++pw) {
                const int v = row[pw];
                cnt += ((v == 2) || (v == 3)) ? 1 : 0;
            }
        }
        patchCnt[p] = cnt;
    }
    __syncthreads();

    // pm >= 0.5  <=>  count >= 128 ; bit position == (hw & 31) == lane id
    if (tid < NWORDS) {
        unsigned bits = 0u;
        for (int bi = 0; bi < 32; ++bi)
            if (patchCnt[tid * 32 + bi] >= 128) bits |= (1u << bi);
        fm[tid] = bits;
    }
    __syncthreads();

    if (tid == 0) {
        int cloth = 0;
        for (int k = 0; k < NWORDS; ++k) cloth += __popc(fm[k]);
        for (int k = 0; k < NWORDS; ++k) {
            idw[b * NWORDS + k] = ~fm[k];                               // id_mask
            clw[b * NWORDS + k] = (cloth == 0) ? 0xFFFFFFFFu : fm[k];   // fallback
        }
        idc[b] = fmaxf((float)(HW - cloth), 1.0f);                      // max(cnt,1)
        clc[b] = (cloth == 0) ? (float)HW : (float)cloth;
    }
}

// ------------------------------ Phase 2 ------------------------------------
__global__ void cmp_pool_kernel(const float* __restrict__ x,
                                const unsigned* __restrict__ idw,
                                const unsigned* __restrict__ clw,
                                const float* __restrict__ idc,
                                const float* __restrict__ clc,
                                float* __restrict__ out) {
    __shared__ float tile[CH_PER_BLOCK * HW];       // 36864 bytes

    const int b    = blockIdx.x >> 7;               // 128 tiles of 32 ch per batch
    const int tle  = blockIdx.x & 127;
    const int c0   = tle * CH_PER_BLOCK;
    const int tid  = threadIdx.x;
    const int wave = tid >> 5;
    const int lane = tid & 31;

    // This wave's 4 channels: contiguous 4608-byte region of x and of the tile.
    const int cw = c0 + wave * 4;
    const char* wsrc = (const char*)(x + ((size_t)b * CHAN + cw) * HW);

    // LDS byte offset: generic LDS pointers carry the WG-relative offset in
    // the low 32 bits (aperture in [63:48]) -> exactly the async VDST operand.
    const unsigned wbase = (unsigned)(uintptr_t)&tile[wave * 4 * HW];

    // Per-wave async staging: 288 b128 chunks, 9 per lane (lane L -> L+32i).
    // NT hint: x is a read-once 302 MB stream, bypass L2 retention.
#pragma unroll
    for (int i = 0; i < 9; ++i) {
        const int chunk   = lane + 32 * i;
        const unsigned lo = wbase + (unsigned)chunk * 16u;
        const char* gp    = wsrc + (size_t)chunk * 16u;
        asm volatile("global_load_async_to_lds_b128 %0, %1, off th:TH_LOAD_NT"
                     :: "v"(lo), "v"(gp) : "memory");
    }
    // Wait only on THIS wave's async transfers -- no __syncthreads needed.
    asm volatile("s_wait_asynccnt 0" ::: "memory");

#pragma unroll
    for (int j = 0; j < 4; ++j) {
        const int lc = wave * 4 + j;
        const int c  = c0 + lc;
        const bool isCloth = (c >= CHAN / 2);       // uniform per block
        const unsigned* words = (isCloth ? clw : idw) + b * NWORDS;
        const float cnt = isCloth ? clc[b] : idc[b];

        float mx = -INFINITY;
        float sm = 0.0f;
#pragma unroll
        for (int k = 0; k < NWORDS; ++k) {          // lane L covers hw = 32k+L
            const float v = tile[lc * HW + k * 32 + lane];  // bank-conflict free
            if ((words[k] >> lane) & 1u) {
                mx = fmaxf(mx, v);
                sm += v;
            }
        }
        // wave32 butterfly reduction
#pragma unroll
        for (int off = 16; off > 0; off >>= 1) {
            mx = fmaxf(mx, __shfl_xor(mx, off, 32));
            sm += __shfl_xor(sm, off, 32);
        }
        if (lane == 0) {
            float* op = out + (isCloth ? (size_t)BATCH * CHAN : 0);
            const int cl = c & (CHAN / 2 - 1);
            op[(size_t)b * CHAN + cl]              = mx;        // masked max
            op[(size_t)b * CHAN + (CHAN / 2) + cl] = sm / cnt;  // masked mean
        }
    }
}

// ------------------------------ Launch -------------------------------------
extern "C" void kernel_launch(void* const* d_in, const int* in_sizes, int n_in,
                              void* d_out, int out_size, void* d_ws, size_t ws_size,
                              hipStream_t stream) {
    (void)in_sizes; (void)n_in; (void)out_size; (void)ws_size;
    const float* x   = (const float*)d_in[0];
    const int* mask  = (const int*)d_in[1];
    float* out       = (float*)d_out;

    unsigned* idw = (unsigned*)d_ws;                 // [64][9]
    unsigned* clw = idw + BATCH * NWORDS;            // [64][9]
    float* idc    = (float*)(clw + BATCH * NWORDS);  // [64]
    float* clc    = idc + BATCH;                     // [64]

    cmp_mask_words_kernel<<<BATCH, 256, 0, stream>>>(mask, idw, clw, idc, clc);
    cmp_pool_kernel<<<BATCH * (CHAN / CH_PER_BLOCK), 256, 0, stream>>>(
        x, idw, clw, idc, clc, out);
}